// GNNEncoder_29506425323886
// MI455X (gfx1250) — compile-verified
//
#include <hip/hip_runtime.h>

// Problem constants (match reference)
#define NU_N   200000
#define NE_N   50000
#define E_N    2000000
#define FU_K   256
#define H_K    128
#define L_N    2
#define BN_EPS 1e-5f

typedef float v2f __attribute__((ext_vector_type(2)));
typedef float v8f __attribute__((ext_vector_type(8)));

// ---------------------------------------------------------------- utilities
__global__ __launch_bounds__(256) void zero_kernel(float* __restrict__ p, int n) {
    int i = blockIdx.x * 256 + threadIdx.x;
    int stride = gridDim.x * 256;
    for (; i < n; i += stride) p[i] = 0.0f;
}

// Degrees as float (exact for small integer counts); computed once, reused.
__global__ __launch_bounds__(256) void degree_kernel(const int* __restrict__ eu,
                                                     const int* __restrict__ ee,
                                                     float* __restrict__ cu,
                                                     float* __restrict__ ce) {
    int i = blockIdx.x * 256 + threadIdx.x;
    if (i < E_N) {
        atomicAdd(&cu[eu[i]], 1.0f);
        atomicAdd(&ce[ee[i]], 1.0f);
    }
}

// One warp (32 lanes) per edge; each lane owns 4 contiguous features (float4
// load, 4 global_atomic_add_f32). H=128 = 32 lanes * 4.
__global__ __launch_bounds__(256) void scatter_kernel(const float* __restrict__ xsrc,
                                                      const int* __restrict__ sidx,
                                                      const int* __restrict__ didx,
                                                      float* __restrict__ acc) {
    int gtid = blockIdx.x * 256 + threadIdx.x;
    int e    = gtid >> 5;
    int lane = gtid & 31;
    if (e >= E_N) return;
    int s = sidx[e];
    int d = didx[e];
    float4 v = *(const float4*)(xsrc + (size_t)s * H_K + lane * 4);
    float* dst = acc + (size_t)d * H_K + lane * 4;
    atomicAdd(dst + 0, v.x);
    atomicAdd(dst + 1, v.y);
    atomicAdd(dst + 2, v.z);
    atomicAdd(dst + 3, v.w);
}

// ------------------------------------------------- WMMA fp32 GEMM kernels
// V_WMMA_F32_16X16X4_F32 fragment mapping (ISA 7.12.2):
//   A 16x4 : lane -> M = lane&15 ; VGPR v holds k = v + 2*(lane>=16)
//   B 4x16 : lane -> N = lane&15 ; VGPR v holds k = v + 2*(lane>=16)
//   C/D    : VGPR r -> M = r + 8*(lane>=16) ; N = lane&15

// Y[M x 128] = X[M x 256] @ W[128 x 256]^T + bias   (initial projections)
__global__ __launch_bounds__(256) void proj_gemm_kernel(const float* __restrict__ X,
                                                        const float* __restrict__ W,
                                                        const float* __restrict__ bias,
                                                        float* __restrict__ Y) {
    __shared__ float As[16][FU_K + 4];           // padded: avoid bank conflicts
    const int tid  = threadIdx.x;
    const size_t row0 = (size_t)blockIdx.x * 16;

    // Stage A tile: 16 rows x 256 cols, fully coalesced (thread = column).
    for (int i = 0; i < 16; ++i)
        As[i][tid] = X[(row0 + i) * FU_K + tid];
    __syncthreads();

    const int wave = tid >> 5;
    const int lane = tid & 31;
    const int hh   = lane >> 4;        // half-wave select
    const int ml   = lane & 15;
    const int n    = wave * 16 + ml;   // output column this lane covers in B/C

    v8f c = {0.f, 0.f, 0.f, 0.f, 0.f, 0.f, 0.f, 0.f};
    for (int k0 = 0; k0 < FU_K; k0 += 4) {
        const int kk = k0 + 2 * hh;
        v2f a, b;
        a.x = As[ml][kk];
        a.y = As[ml][kk + 1];
        b.x = W[(size_t)n * FU_K + kk];
        b.y = W[(size_t)n * FU_K + kk + 1];
        c = __builtin_amdgcn_wmma_f32_16x16x4_f32(false, a, false, b,
                                                  (short)0, c, false, false);
    }
    const float bn = bias[n];
    for (int r = 0; r < 8; ++r) {
        size_t row = row0 + r + 8 * hh;
        Y[row * H_K + n] = c[r] + bn;
    }
}

// Y = (Acc/max(cnt,1)) @ Wl^T + bl + Xd @ Wr^T, written in place over Acc.
// Safe in place: the block stages its 16 exclusive rows into LDS first.
__global__ __launch_bounds__(256) void sage_gemm_kernel(const float* __restrict__ Acc,
                                                        const float* __restrict__ Xd,
                                                        const float* __restrict__ Wl,
                                                        const float* __restrict__ Wr,
                                                        const float* __restrict__ bl,
                                                        const float* __restrict__ cnt,
                                                        float* __restrict__ Y) {
    __shared__ float A1[16][H_K + 4];
    __shared__ float A2[16][H_K + 4];
    const int tid  = threadIdx.x;
    const size_t row0 = (size_t)blockIdx.x * 16;

    for (int i = 0; i < 8; ++i) {                 // 2048 elems / 256 threads
        int flat = i * 256 + tid;
        int rr = flat >> 7;                       // row in tile
        int cc = flat & 127;                      // col
        size_t grow = row0 + rr;
        float inv = 1.0f / fmaxf(cnt[grow], 1.0f);
        A1[rr][cc] = Acc[grow * H_K + cc] * inv;  // mean aggregation
        A2[rr][cc] = Xd[grow * H_K + cc];
    }
    __syncthreads();

    const int wave = tid >> 5;
    const int lane = tid & 31;
    const int hh   = lane >> 4;
    const int ml   = lane & 15;
    const int n    = wave * 16 + ml;

    v8f c = {0.f, 0.f, 0.f, 0.f, 0.f, 0.f, 0.f, 0.f};
    for (int k0 = 0; k0 < H_K; k0 += 4) {
        const int kk = k0 + 2 * hh;
        v2f a1, b1, a2, b2;
        a1.x = A1[ml][kk];
        a1.y = A1[ml][kk + 1];
        b1.x = Wl[(size_t)n * H_K + kk];
        b1.y = Wl[(size_t)n * H_K + kk + 1];
        c = __builtin_amdgcn_wmma_f32_16x16x4_f32(false, a1, false, b1,
                                                  (short)0, c, false, false);
        a2.x = A2[ml][kk];
        a2.y = A2[ml][kk + 1];
        b2.x = Wr[(size_t)n * H_K + kk];
        b2.y = Wr[(size_t)n * H_K + kk + 1];
        c = __builtin_amdgcn_wmma_f32_16x16x4_f32(false, a2, false, b2,
                                                  (short)0, c, false, false);
    }
    const float bn = bl[n];
    for (int r = 0; r < 8; ++r) {
        size_t row = row0 + r + 8 * hh;
        Y[row * H_K + n] = c[r] + bn;
    }
}

// ---------------------------------------------------------------- batchnorm
#define BN_ROWS 512
__global__ __launch_bounds__(256) void bn_stats_kernel(const float* __restrict__ Y,
                                                       float* __restrict__ sum,
                                                       float* __restrict__ sumsq,
                                                       int M) {
    const int col = threadIdx.x & 127;
    const int rs  = threadIdx.x >> 7;             // 0 or 1
    int rbeg = blockIdx.x * BN_ROWS + rs;
    int rend = min(M, (int)(blockIdx.x + 1) * BN_ROWS);
    float s = 0.f, q = 0.f;
    for (int r = rbeg; r < rend; r += 2) {
        float v = Y[(size_t)r * H_K + col];
        s += v;
        q += v * v;
    }
    atomicAdd(&sum[col], s);
    atomicAdd(&sumsq[col], q);
}

__global__ __launch_bounds__(128) void bn_finalize_kernel(const float* __restrict__ sum,
                                                          const float* __restrict__ sumsq,
                                                          const float* __restrict__ g,
                                                          const float* __restrict__ beta,
                                                          float* __restrict__ scale,
                                                          float* __restrict__ shift,
                                                          float invM) {
    int c = threadIdx.x;
    float m   = sum[c] * invM;
    float var = sumsq[c] * invM - m * m;          // biased variance (matches ref)
    float a   = g[c] * rsqrtf(var + BN_EPS);
    scale[c] = a;
    shift[c] = beta[c] - m * a;
}

__global__ __launch_bounds__(256) void bn_apply_kernel(const float* __restrict__ Y,
                                                       const float* __restrict__ scale,
                                                       const float* __restrict__ shift,
                                                       float* __restrict__ X, int total) {
    int i = blockIdx.x * 256 + threadIdx.x;
    int stride = gridDim.x * 256;
    for (; i < total; i += stride) {
        int col = i & 127;
        X[i] = fmaxf(Y[i] * scale[col] + shift[col], 0.0f);
    }
}

// ---------------------------------------------------------------- launcher
extern "C" void kernel_launch(void* const* d_in, const int* in_sizes, int n_in,
                              void* d_out, int out_size, void* d_ws, size_t ws_size,
                              hipStream_t stream) {
    const float* x_user  = (const float*)d_in[0];
    const float* x_event = (const float*)d_in[1];
    const int*   e_user  = (const int*)d_in[2];
    const int*   e_event = (const int*)d_in[3];
    const float* upW = (const float*)d_in[4];
    const float* upb = (const float*)d_in[5];
    const float* epW = (const float*)d_in[6];
    const float* epb = (const float*)d_in[7];
    const float* uWl = (const float*)d_in[8];
    const float* ubl = (const float*)d_in[9];
    const float* uWr = (const float*)d_in[10];
    const float* ubg = (const float*)d_in[11];
    const float* ubb = (const float*)d_in[12];
    const float* eWl = (const float*)d_in[13];
    const float* ebl = (const float*)d_in[14];
    const float* eWr = (const float*)d_in[15];
    const float* ebg = (const float*)d_in[16];
    const float* ebb = (const float*)d_in[17];

    // d_out doubles as the xu/xe working buffers (final values land there).
    float* xu = (float*)d_out;                    // NU*H
    float* xe = (float*)d_out + (size_t)NU_N * H_K;

    // workspace layout
    float* acc     = (float*)d_ws;                // NU*H (largest side)
    float* cnt_u   = acc + (size_t)NU_N * H_K;
    float* cnt_e   = cnt_u + NU_N;
    float* s_sum   = cnt_e + NE_N;
    float* s_sq    = s_sum + H_K;
    float* s_scale = s_sq + H_K;
    float* s_shift = s_scale + H_K;

    const int ZGRID = 4096;
    const int scatter_blocks = (E_N * 32) / 256;  // 1 warp per edge

    // degrees (once; shared by both layers)
    zero_kernel<<<ZGRID, 256, 0, stream>>>(cnt_u, NU_N);
    zero_kernel<<<ZGRID, 256, 0, stream>>>(cnt_e, NE_N);
    degree_kernel<<<(E_N + 255) / 256, 256, 0, stream>>>(e_user, e_event, cnt_u, cnt_e);

    // initial projections (WMMA)
    proj_gemm_kernel<<<NU_N / 16, 256, 0, stream>>>(x_user, upW, upb, xu);
    proj_gemm_kernel<<<NE_N / 16, 256, 0, stream>>>(x_event, epW, epb, xe);

    for (int i = 0; i < L_N; ++i) {
        // ---- event -> user
        zero_kernel<<<ZGRID, 256, 0, stream>>>(acc, NU_N * H_K);
        scatter_kernel<<<scatter_blocks, 256, 0, stream>>>(xe, e_event, e_user, acc);
        sage_gemm_kernel<<<NU_N / 16, 256, 0, stream>>>(
            acc, xu, uWl + (size_t)i * H_K * H_K, uWr + (size_t)i * H_K * H_K,
            ubl + i * H_K, cnt_u, acc);
        zero_kernel<<<1, 256, 0, stream>>>(s_sum, 2 * H_K);   // sum+sumsq contiguous
        bn_stats_kernel<<<(NU_N + BN_ROWS - 1) / BN_ROWS, 256, 0, stream>>>(acc, s_sum, s_sq, NU_N);
        bn_finalize_kernel<<<1, 128, 0, stream>>>(s_sum, s_sq, ubg + i * H_K, ubb + i * H_K,
                                                  s_scale, s_shift, 1.0f / NU_N);
        bn_apply_kernel<<<ZGRID, 256, 0, stream>>>(acc, s_scale, s_shift, xu, NU_N * H_K);

        // ---- user -> event
        zero_kernel<<<ZGRID, 256, 0, stream>>>(acc, NE_N * H_K);
        scatter_kernel<<<scatter_blocks, 256, 0, stream>>>(xu, e_user, e_event, acc);
        sage_gemm_kernel<<<NE_N / 16, 256, 0, stream>>>(
            acc, xe, eWl + (size_t)i * H_K * H_K, eWr + (size_t)i * H_K * H_K,
            ebl + i * H_K, cnt_e, acc);
        zero_kernel<<<1, 256, 0, stream>>>(s_sum, 2 * H_K);
        bn_stats_kernel<<<(NE_N + BN_ROWS - 1) / BN_ROWS, 256, 0, stream>>>(acc, s_sum, s_sq, NE_N);
        bn_finalize_kernel<<<1, 128, 0, stream>>>(s_sum, s_sq, ebg + i * H_K, ebb + i * H_K,
                                                  s_scale, s_shift, 1.0f / NE_N);
        bn_apply_kernel<<<ZGRID, 256, 0, stream>>>(acc, s_scale, s_shift, xe, NE_N * H_K);
    }
}